// MeanIOU_10685878633048
// MI455X (gfx1250) — compile-verified
//
#include <hip/hip_runtime.h>
#include <hip/hip_bf16.h>
#include <stdint.h>

// Mean IoU: per-pixel argmax over 21 classes -> 21x21 confusion matrix -> IoU scalar.
// Memory-bound: 184 MB streamed once; floor ~8us @ 23.3 TB/s (RT policy kept so the
// working set can live in the 192 MB L2 across graph replays).
// CDNA5 path: double-buffered GLOBAL_LOAD_ASYNC_TO_LDS_B128 staging of 4 KB class
// tiles, synchronized per-wave with s_wait_asynccnt (no workgroup barriers in the
// hot loop: each thread consumes exactly the 16 LDS bytes it staged).

#define NUM_CLASSES 21
#define CONF_BINS   (NUM_CLASSES * NUM_CLASSES)
#define WH          (512 * 512)            // pixels per image, = 2^18
#define PIX_PER_BLK 1024
#define THREADS     256

#if __has_builtin(__builtin_amdgcn_s_wait_asynccnt)
#define WAIT_ASYNCCNT(n) __builtin_amdgcn_s_wait_asynccnt(n)
#else
#define WAIT_ASYNCCNT(n) asm volatile("s_wait_asynccnt %0" :: "i"(n) : "memory")
#endif

// One 16-byte-per-lane async DMA: global -> LDS (512 B per wave, coalesced).
// vdst = LDS byte address (low 32 bits of the generic pointer: LDS aperture
// keeps the LDS offset in addr[31:0]), vaddr = 64-bit global address, GV mode.
__device__ __forceinline__ void async_b128_to_lds(const void* gaddr, void* ldsaddr) {
    uint64_t ga = (uint64_t)(uintptr_t)gaddr;
    uint32_t la = (uint32_t)(uintptr_t)ldsaddr;
    asm volatile("global_load_async_to_lds_b128 %0, %1, off"
                 :
                 : "v"(la), "v"(ga)
                 : "memory");
}

__global__ __launch_bounds__(THREADS) void miou_conf_kernel(
    const float* __restrict__ logits,
    const int*   __restrict__ labels,
    unsigned int* __restrict__ gconf)
{
    __shared__ float        cls[2][PIX_PER_BLK];      // double-buffered class tile (8 KB)
    __shared__ unsigned int hist[CONF_BINS];          // per-block confusion matrix

    const int tid = threadIdx.x;
    for (int i = tid; i < CONF_BINS; i += THREADS) hist[i] = 0u;

    const unsigned int blockStart = blockIdx.x * PIX_PER_BLK;  // global pixel index
    const unsigned int b     = blockStart >> 18;               // / WH
    const unsigned int inImg = blockStart & (WH - 1);          // % WH
    // element offset of this thread's 4 pixels for class 0
    const uint64_t row0 = (uint64_t)(b * NUM_CLASSES) * WH + inImg + 4u * tid;

    float* myLds0 = &cls[0][4 * tid];
    float* myLds1 = &cls[1][4 * tid];

    // prologue: two class tiles in flight
    async_b128_to_lds(logits + row0,                    myLds0);   // class 0
    async_b128_to_lds(logits + row0 + (uint64_t)WH,     myLds1);   // class 1

    // labels load overlaps the async pipeline
    const int4 lab4 = *(const int4*)(labels + blockStart + 4 * tid);

    // ---- c = 0: initialize argmax state ----
    WAIT_ASYNCCNT(1);                       // class 0 resident (in-order completion)
    float4 v = *(const float4*)myLds0;
    float best0 = v.x, best1 = v.y, best2 = v.z, best3 = v.w;
    int   bi0 = 0, bi1 = 0, bi2 = 0, bi3 = 0;

    // ---- steady state c = 1..19: branch-free issue/wait/consume ----
#pragma unroll
    for (int c = 1; c < NUM_CLASSES - 1; ++c) {
        async_b128_to_lds(logits + row0 + (uint64_t)(c + 1) * WH,
                          ((c + 1) & 1) ? myLds1 : myLds0);
        WAIT_ASYNCCNT(1);                   // class c resident
        v = *(const float4*)((c & 1) ? myLds1 : myLds0);
        // strict > keeps first-occurrence argmax (jnp.argmax semantics)
        if (v.x > best0) { best0 = v.x; bi0 = c; }
        if (v.y > best1) { best1 = v.y; bi1 = c; }
        if (v.z > best2) { best2 = v.z; bi2 = c; }
        if (v.w > best3) { best3 = v.w; bi3 = c; }
    }

    // ---- epilogue c = 20 (20 & 1 == 0 -> buffer 0) ----
    WAIT_ASYNCCNT(0);
    v = *(const float4*)myLds0;
    if (v.x > best0) { best0 = v.x; bi0 = NUM_CLASSES - 1; }
    if (v.y > best1) { best1 = v.y; bi1 = NUM_CLASSES - 1; }
    if (v.z > best2) { best2 = v.z; bi2 = NUM_CLASSES - 1; }
    if (v.w > best3) { best3 = v.w; bi3 = NUM_CLASSES - 1; }

    __syncthreads();   // hist zeroing (top) must precede the atomics below

    atomicAdd(&hist[lab4.x * NUM_CLASSES + bi0], 1u);
    atomicAdd(&hist[lab4.y * NUM_CLASSES + bi1], 1u);
    atomicAdd(&hist[lab4.z * NUM_CLASSES + bi2], 1u);
    atomicAdd(&hist[lab4.w * NUM_CLASSES + bi3], 1u);

    __syncthreads();

    for (int i = tid; i < CONF_BINS; i += THREADS) {
        unsigned int h = hist[i];
        if (h) atomicAdd(&gconf[i], h);
    }
}

__global__ void miou_zero_kernel(unsigned int* __restrict__ gconf) {
    int i = blockIdx.x * blockDim.x + threadIdx.x;
    if (i < CONF_BINS) gconf[i] = 0u;
}

__global__ void miou_finalize_kernel(const unsigned int* __restrict__ gconf,
                                     float* __restrict__ out)
{
    const int c = threadIdx.x;   // one wave32; lanes 21..31 contribute zeros
    float iou = 0.f, valid = 0.f;
    if (c < NUM_CLASSES) {
        const float inter = (float)gconf[c * NUM_CLASSES + c];
        float rowsum = 0.f, colsum = 0.f;
        for (int j = 0; j < NUM_CLASSES; ++j) {
            rowsum += (float)gconf[c * NUM_CLASSES + j];   // label_count
            colsum += (float)gconf[j * NUM_CLASSES + c];   // pred_count
        }
        const float uni = rowsum + colsum - inter;
        const float vv = (inter + 1e-9f) / (uni + 1e-9f);
        if (rowsum > 0.f) { iou = vv; valid = 1.f; }
    }
    for (int off = 16; off > 0; off >>= 1) {
        iou   += __shfl_down(iou,   off, 32);
        valid += __shfl_down(valid, off, 32);
    }
    if (c == 0) out[0] = iou / fmaxf(valid, 1.0f);
}

extern "C" void kernel_launch(void* const* d_in, const int* in_sizes, int n_in,
                              void* d_out, int out_size, void* d_ws, size_t ws_size,
                              hipStream_t stream)
{
    const float* logits = (const float*)d_in[0];   // [8, 21, 512, 512] f32
    const int*   labels = (const int*)d_in[1];     // [8, 512, 512] i32
    unsigned int* gconf = (unsigned int*)d_ws;     // 441 u32 counters
    float* out = (float*)d_out;                    // scalar

    miou_zero_kernel<<<2, 256, 0, stream>>>(gconf);

    const int totalPix = in_sizes[1];              // B*W*H = 2,097,152
    const int blocks   = totalPix / PIX_PER_BLK;   // 2048
    miou_conf_kernel<<<blocks, THREADS, 0, stream>>>(logits, labels, gconf);

    miou_finalize_kernel<<<1, 32, 0, stream>>>(gconf, out);
}